// channel_clustering_loss_1958505087268
// MI455X (gfx1250) — compile-verified
//
#include <hip/hip_runtime.h>

// Shapes (compile-time from reference): feature [16, 32, 48, 48, 48] f32
#define Bn   16
#define Cn   32
#define Xn   48
#define Yn   48
#define Zn   48
#define YZn  (Yn * Zn)          // 2304
#define XYn  (Xn * Yn)          // 2304 lines per (b,c)
#define XYZn (Xn * Yn * Zn)     // 110592
#define BCn  (Bn * Cn)          // 512
#define SUM_NUM_D 56623104.0    // B*C*X*Y*Z

typedef __attribute__((ext_vector_type(2))) float v2f;
typedef __attribute__((ext_vector_type(8))) float v8f;

// Reduce 64 f32 values held 2-per-lane across a full wave32 using
// V_WMMA_F32_16X16X4_F32 with an all-ones B matrix:
//   D[m][n] = sum_k A[m][k]  (row sums, replicated over n)
// Lane layout (ISA 7.12.2): lanes 0-15 hold rows m=lane K0,K1; lanes 16-31
// hold rows m=lane-16 K2,K3. D VGPR j: lanes 0-15 -> rowsum(j),
// lanes 16-31 -> rowsum(j+8). Sum of 8 D regs + one shuffle-down by 16 = total.
// WMMA is a fixed hardware FMA tree -> bitwise deterministic.
// EXEC must be all ones: call only from wave-uniform full-wave context.
__device__ __forceinline__ float wave_sum64_wmma(float a0, float a1) {
    v2f A;  A[0] = a0;   A[1] = a1;
    v2f Bm; Bm[0] = 1.0f; Bm[1] = 1.0f;
    v8f Cz = {0.f, 0.f, 0.f, 0.f, 0.f, 0.f, 0.f, 0.f};
    v8f D = __builtin_amdgcn_wmma_f32_16x16x4_f32(
        /*neg_a=*/false, A, /*neg_b=*/false, Bm,
        /*c_mod=*/(short)0, Cz, /*reuse_a=*/false, /*reuse_b=*/false);
    float r = ((D[0] + D[1]) + (D[2] + D[3])) + ((D[4] + D[5]) + (D[6] + D[7]));
    r += __shfl_down(r, 16, 32);   // lane0: rows 0-7 + rows 8-15 = total
    return r;                      // valid in lanes 0..15
}

// ---------------------------------------------------------------------------
// Kernel 1: per-(b,c) statistics in one streaming read of the slab.
//   sums[bc]      = sum of values (for global mean)
//   Sx[bc][x]     = sum over (y,z) of value^2
//   Sy[bc][y]     = sum over (x,z) of value^2
//   amIdx[bc]     = flat argmax (first occurrence == strict '>' scan)
// Deterministic: line sums staged in LDS, fixed-order tree reductions.
// ---------------------------------------------------------------------------
__global__ __launch_bounds__(256) void stats_kernel(
        const float* __restrict__ f,
        float* __restrict__ sums, float* __restrict__ Sx,
        float* __restrict__ Sy, int* __restrict__ amIdx) {
    __shared__ float line2[XYn];   // per-line sum of squares
    __shared__ float redA[256];
    __shared__ int   redI[256];
    const int tid = threadIdx.x;
    const int bc  = blockIdx.x;
    const float* base = f + (size_t)bc * XYZn;

    float ts1  = 0.0f;
    float mmax = -__builtin_huge_valf();
    int   midx = 0;

    for (int li = tid; li < XYn; li += 256) {              // 9 lines/thread
        const float4* lp = (const float4*)(base + li * Zn);
        if (li + 256 < XYn)
            __builtin_prefetch(base + (li + 256) * Zn, 0, 0);  // global_prefetch_b8
        float s1 = 0.0f, s2 = 0.0f;
        #pragma unroll
        for (int q = 0; q < Zn / 4; ++q) {                 // 12 x float4
            float4 v = lp[q];
            const int zb = li * Zn + q * 4;
            if (v.x > mmax) { mmax = v.x; midx = zb + 0; }
            if (v.y > mmax) { mmax = v.y; midx = zb + 1; }
            if (v.z > mmax) { mmax = v.z; midx = zb + 2; }
            if (v.w > mmax) { mmax = v.w; midx = zb + 3; }
            s1 += (v.x + v.y) + (v.z + v.w);
            s2 += (v.x * v.x + v.y * v.y) + (v.z * v.z + v.w * v.w);
        }
        ts1 += s1;
        line2[li] = s2;
    }
    __syncthreads();

    if (tid < Xn) {                       // Sx[x] = sum_y line2[x][y]
        float s = 0.0f;
        for (int y = 0; y < Yn; ++y) s += line2[tid * Yn + y];
        Sx[bc * Xn + tid] = s;
    }
    if (tid >= 64 && tid < 64 + Yn) {     // Sy[y] = sum_x line2[x][y]
        const int y = tid - 64;
        float s = 0.0f;
        for (int x = 0; x < Xn; ++x) s += line2[x * Yn + y];
        Sy[bc * Yn + y] = s;
    }

    redA[tid] = ts1;
    __syncthreads();
    for (int off = 128; off > 0; off >>= 1) {
        if (tid < off) redA[tid] += redA[tid + off];
        __syncthreads();
    }
    if (tid == 0) sums[bc] = redA[0];
    __syncthreads();

    redA[tid] = mmax; redI[tid] = midx;
    __syncthreads();
    for (int off = 128; off > 0; off >>= 1) {
        if (tid < off) {
            const float vo = redA[tid + off];
            const int   io = redI[tid + off];
            if (vo > redA[tid] || (vo == redA[tid] && io < redI[tid])) {
                redA[tid] = vo; redI[tid] = io;
            }
        }
        __syncthreads();
    }
    if (tid == 0) amIdx[bc] = redI[0];
}

// ---------------------------------------------------------------------------
// Kernel 2 (tiny): mgr = mean(feature); dis_loss from separable marginals:
//   dis = sum_bc [ sum_x (mx-x)^2*Sx + sum_y (my-y)^2*Sy + X*Y*sum_z (mz-z)^2 ]
// Double accumulation + deterministic tree.
// ---------------------------------------------------------------------------
__global__ __launch_bounds__(256) void disloss_kernel(
        const float* __restrict__ sums, const float* __restrict__ Sx,
        const float* __restrict__ Sy, const int* __restrict__ amIdx,
        float* __restrict__ mgrOut, float* __restrict__ dout) {
    __shared__ double redD[256];
    const int tid = threadIdx.x;
    double disAcc = 0.0, sAcc = 0.0;

    for (int bc = tid; bc < BCn; bc += 256) {              // 2 bc per thread
        const int am  = amIdx[bc];
        const float mx = (float)(am / YZn);
        const int  rem = am % YZn;
        const float my = (float)(rem / Zn);
        const float mz = (float)(rem % Zn);
        double acc = 0.0;
        for (int x = 0; x < Xn; ++x) {
            const float d = mx - (float)x;
            acc += (double)(d * d) * (double)Sx[bc * Xn + x];
        }
        for (int y = 0; y < Yn; ++y) {
            const float d = my - (float)y;
            acc += (double)(d * d) * (double)Sy[bc * Yn + y];
        }
        double zacc = 0.0;
        for (int z = 0; z < Zn; ++z) {
            const float d = mz - (float)z;
            zacc += (double)(d * d);
        }
        acc += (double)(Xn * Yn) * zacc;   // dz^2 term NOT scaled by r^2 (faithful)
        disAcc += acc;
        sAcc   += (double)sums[bc];
    }

    redD[tid] = disAcc;
    __syncthreads();
    for (int off = 128; off > 0; off >>= 1) {
        if (tid < off) redD[tid] += redD[tid + off];
        __syncthreads();
    }
    if (tid == 0) dout[0] = (float)(redD[0] / SUM_NUM_D);
    __syncthreads();

    redD[tid] = sAcc;
    __syncthreads();
    for (int off = 128; off > 0; off >>= 1) {
        if (tid < off) redD[tid] += redD[tid + off];
        __syncthreads();
    }
    if (tid == 0) *mgrOut = (float)(redD[0] / SUM_NUM_D);
}

// ---------------------------------------------------------------------------
// Kernel 3: div_loss pass. One block = 1024 contiguous voxels of one sample b.
// Blocks mapped in REVERSE so the tail of pass-1 data still resident in the
// 192MB L2 is consumed before eviction (226MB tensor, ~85% potential L2 hits).
// Per voxel: top-2 over 32 channels (first-occurrence argmax), then
// acc += v^2 * (max_others - mgr)^2. Block partial via WMMA wave reduce.
// ---------------------------------------------------------------------------
__global__ __launch_bounds__(256) void divloss_kernel(
        const float* __restrict__ f, const float* __restrict__ mgrP,
        float* __restrict__ partials) {
    __shared__ float red[256];
    const int tid   = threadIdx.x;
    const int rb    = (int)gridDim.x - 1 - (int)blockIdx.x;   // reverse order
    const int b     = rb / (XYZn / 1024);                     // / 108
    const int chunk = rb % (XYZn / 1024);
    const float* fb = f + (size_t)b * Cn * XYZn;
    const float mgr = *mgrP;

    float acc = 0.0f;
    #pragma unroll
    for (int k = 0; k < 4; ++k) {
        const int vox = chunk * 1024 + k * 256 + tid;
        float v[Cn];
        #pragma unroll
        for (int c = 0; c < Cn; ++c) v[c] = fb[(size_t)c * XYZn + vox];

        float m1 = -__builtin_huge_valf(), m2 = -__builtin_huge_valf();
        int ac = -1;
        #pragma unroll
        for (int c = 0; c < Cn; ++c) {         // strict '>' => first occurrence
            if (v[c] > m1)      { m2 = m1; m1 = v[c]; ac = c; }
            else if (v[c] > m2) { m2 = v[c]; }
        }
        #pragma unroll
        for (int c = 0; c < Cn; ++c) {
            const float mo = (c == ac) ? m2 : m1;
            const float d  = mo - mgr;
            acc += (v[c] * v[c]) * (d * d);
        }
    }

    red[tid] = acc;
    __syncthreads();
    if (tid < 128) red[tid] += red[tid + 128];
    __syncthreads();
    if (tid < 64)  red[tid] += red[tid + 64];
    __syncthreads();
    if (tid < 32) {                            // wave-uniform, full wave32
        const float t = wave_sum64_wmma(red[tid], red[tid + 32]);
        if (tid == 0) partials[blockIdx.x] = t;
    }
}

// ---------------------------------------------------------------------------
// Kernel 4: deterministic final reduce of block partials via WMMA.
// ---------------------------------------------------------------------------
__global__ __launch_bounds__(32) void final_div_kernel(
        const float* __restrict__ partials, int n, float* __restrict__ dout) {
    const int lane = threadIdx.x;
    float a0 = 0.0f, a1 = 0.0f;
    for (int i = lane;      i < n; i += 64) a0 += partials[i];
    for (int i = lane + 32; i < n; i += 64) a1 += partials[i];
    const float t = wave_sum64_wmma(a0, a1);
    if (lane == 0) dout[1] = (float)((double)t / SUM_NUM_D);
}

// ---------------------------------------------------------------------------
extern "C" void kernel_launch(void* const* d_in, const int* in_sizes, int n_in,
                              void* d_out, int out_size, void* d_ws, size_t ws_size,
                              hipStream_t stream) {
    (void)in_sizes; (void)n_in; (void)out_size; (void)ws_size;
    const float* f = (const float*)d_in[0];
    float* dout = (float*)d_out;

    // Workspace layout (~208 KB total, all overwritten every call):
    float* ws    = (float*)d_ws;
    float* sums  = ws;                              // 512
    float* Sx    = ws + BCn;                        // 512*48
    float* Sy    = Sx + (size_t)BCn * Xn;           // 512*48
    int*   amI   = (int*)(Sy + (size_t)BCn * Yn);   // 512
    float* mgrP  = (float*)(amI + BCn);             // 1
    float* parts = mgrP + 1;                        // 1728

    const int nDivBlocks = Bn * (XYZn / 1024);      // 1728

    stats_kernel  <<<BCn,        256, 0, stream>>>(f, sums, Sx, Sy, amI);
    disloss_kernel<<<1,          256, 0, stream>>>(sums, Sx, Sy, amI, mgrP, dout);
    divloss_kernel<<<nDivBlocks, 256, 0, stream>>>(f, mgrP, parts);
    final_div_kernel<<<1,         32, 0, stream>>>(parts, nDivBlocks, dout);
}